// LIFLayer_86260123173247
// MI455X (gfx1250) — compile-verified
//
#include <hip/hip_runtime.h>
#include <cstdint>

// ---------------------------------------------------------------------------
// LIF scan, MI455X (gfx1250). Bandwidth-bound streaming recurrence:
//   v_pre = v + K*(I - v); s = (v_pre >= 1) ? 1 : 0; v = v_pre*(1-s)
// Data path: async global->LDS double-buffered input tiles, async LDS->global
// double-buffered output tiles (ASYNCcnt), coalesced 128B transactions.
// Roofline: 537 MB traffic @ 23.3 TB/s ~= 23 us floor; ~0.25 FLOP/byte, so
// the win is entirely in the async/coalesced data path, not matrix ops.
// ---------------------------------------------------------------------------

namespace {
constexpr int   kB = 64;
constexpr int   kF = 512;
constexpr int   kL = 1024;
constexpr int   kN = kB * kF;            // 32768 neurons
constexpr float kK   = 0.2f;
constexpr float kThr = 1.0f;

constexpr int BLOCK  = 128;              // threads/block == neurons/block (4 wave32)
constexpr int T      = 32;               // time steps per chunk
constexpr int CHUNKS = kL / T;           // 32
constexpr int QPR    = T / 4;            // 8 float4 quads per tile row
constexpr int PAD    = T + 4;            // 36 floats = 144B row stride (16B aligned)
constexpr int TILE   = BLOCK * PAD;      // floats per LDS tile
constexpr int NTILES = 2 /*in dbl-buf*/ + 6 /*3 outputs x dbl-buf*/;
constexpr size_t LDS_BYTES = size_t(NTILES) * TILE * sizeof(float);
}

// ---- CDNA5 async global<->LDS path (probe-guarded) -------------------------
#if defined(__has_builtin)
#if __has_builtin(__builtin_amdgcn_global_load_async_to_lds_b128) && \
    __has_builtin(__builtin_amdgcn_global_store_async_from_lds_b128) && \
    __has_builtin(__builtin_amdgcn_s_wait_asynccnt)
#define LIF_USE_ASYNC 1
#endif
#endif
#ifndef LIF_USE_ASYNC
#define LIF_USE_ASYNC 0
#endif

#if LIF_USE_ASYNC
// Diagnostic-derived parameter types: pointer to int4 in AS1 (global) for the
// memory side; LDS side assumed int4 in AS3.
typedef int lif_v4i __attribute__((vector_size(4 * sizeof(int))));
typedef __attribute__((address_space(1))) lif_v4i lif_gv4i;
typedef __attribute__((address_space(3))) lif_v4i lif_lv4i;

__device__ __forceinline__ void lif_async_load16(const float* g, float* l) {
  // emits: global_load_async_to_lds_b128 (tracked with ASYNCcnt)
  __builtin_amdgcn_global_load_async_to_lds_b128((lif_gv4i*)g, (lif_lv4i*)l, 0, 0);
}
__device__ __forceinline__ void lif_async_store16(float* g, const float* l) {
  // emits: global_store_async_from_lds_b128 (tracked with ASYNCcnt)
  __builtin_amdgcn_global_store_async_from_lds_b128((lif_gv4i*)g, (lif_lv4i*)l, 0, 0);
}
__device__ __forceinline__ void lif_wait_async_le8() {
  __builtin_amdgcn_s_wait_asynccnt(8);
}
__device__ __forceinline__ void lif_wait_async_0() {
  __builtin_amdgcn_s_wait_asynccnt(0);
}
#else
// Synchronous fallback: keeps identical tiling/correctness, no ASYNCcnt.
__device__ __forceinline__ void lif_async_load16(const float* g, float* l) {
  *(float4*)l = *(const float4*)g;
}
__device__ __forceinline__ void lif_async_store16(float* g, const float* l) {
  *(float4*)g = *(const float4*)l;
}
__device__ __forceinline__ void lif_wait_async_le8() {}
__device__ __forceinline__ void lif_wait_async_0()  {}
#endif

// ---- tile movers -----------------------------------------------------------
// Tile = 128 rows (neurons) x 32 floats (time), row stride PAD floats in LDS.
// j = tid + k*BLOCK enumerates 1024 quads; consecutive lanes hit consecutive
// 16B quads -> each wave instruction covers 4 full 128B cachelines.
__device__ __forceinline__ void lif_issue_tile_load(const float* gbase,
                                                    float* tile, int tid) {
#pragma unroll
  for (int k = 0; k < QPR; ++k) {
    const int j = tid + k * BLOCK;
    const int r = j >> 3;            // j / QPR
    const int q = j & (QPR - 1);
    lif_async_load16(gbase + size_t(r) * kL + q * 4, tile + r * PAD + q * 4);
  }
}

__device__ __forceinline__ void lif_issue_tile_store(float* gbase,
                                                     const float* tile, int tid) {
#pragma unroll
  for (int k = 0; k < QPR; ++k) {
    const int j = tid + k * BLOCK;
    const int r = j >> 3;
    const int q = j & (QPR - 1);
    lif_async_store16(gbase + size_t(r) * kL + q * 4, tile + r * PAD + q * 4);
  }
}

// ---- kernel ----------------------------------------------------------------
__global__ __launch_bounds__(BLOCK, 2)
void LIFLayer_86260123173247_kernel(const float* __restrict__ I,
                                    float* __restrict__ vpre,
                                    float* __restrict__ vnew,
                                    float* __restrict__ spk) {
  extern __shared__ float lds[];
  float* in_tile0 = lds;                       // [2][TILE]
  float* out_tile0 = lds + 2 * TILE;           // [3][2][TILE]

  const int tid = threadIdx.x;
  const int n0  = blockIdx.x * BLOCK;

  // Prologue: async-load chunk 0 into buffer 0.
  lif_issue_tile_load(I + size_t(n0) * kL, in_tile0, tid);

  float v = 0.0f;  // membrane potential, persistent across all 1024 steps

#pragma unroll 1
  for (int c = 0; c < CHUNKS; ++c) {
    const int cur = c & 1;
    float* in_cur  = in_tile0 + cur * TILE;
    float* vp_cur  = out_tile0 + (0 * 2 + cur) * TILE;
    float* vn_cur  = out_tile0 + (1 * 2 + cur) * TILE;
    float* sp_cur  = out_tile0 + (2 * 2 + cur) * TILE;

    if (c + 1 < CHUNKS) {
      // Prefetch next chunk into the other input buffer, then wait until at
      // most those 8 ops remain: guarantees loads(c) landed and stores(c-2)
      // (same output buffers as this chunk) drained.
      lif_issue_tile_load(I + size_t(n0) * kL + (c + 1) * T,
                          in_tile0 + (cur ^ 1) * TILE, tid);
      lif_wait_async_le8();
    } else {
      lif_wait_async_0();
    }
    __syncthreads();   // all waves' loads(c) visible in LDS

    // ---- compute 32 sequential steps for this thread's neuron ----
    const float4* irow = (const float4*)(in_cur + tid * PAD);
    float4* vprow = (float4*)(vp_cur + tid * PAD);
    float4* vnrow = (float4*)(vn_cur + tid * PAD);
    float4* sprow = (float4*)(sp_cur + tid * PAD);

#pragma unroll
    for (int q = 0; q < QPR; ++q) {
      const float4 i4 = irow[q];
      float4 vp4, vn4, s4;
#define LIF_STEP(IN, VP, VN, SP)                         \
      {                                                  \
        const float vpre_ = fmaf(kK, (IN) - v, v);       \
        const float s_    = (vpre_ >= kThr) ? 1.0f : 0.0f; \
        (VP) = vpre_;                                    \
        (SP) = s_;                                       \
        v    = vpre_ * (1.0f - s_);                      \
        (VN) = v;                                        \
      }
      LIF_STEP(i4.x, vp4.x, vn4.x, s4.x)
      LIF_STEP(i4.y, vp4.y, vn4.y, s4.y)
      LIF_STEP(i4.z, vp4.z, vn4.z, s4.z)
      LIF_STEP(i4.w, vp4.w, vn4.w, s4.w)
#undef LIF_STEP
      vprow[q] = vp4;
      vnrow[q] = vn4;
      sprow[q] = s4;
    }

    __syncthreads();   // all waves' result-tile writes visible

    // Stream this chunk's three tiles out (24 async ops / thread).
    lif_issue_tile_store(vpre + size_t(n0) * kL + c * T, vp_cur, tid);
    lif_issue_tile_store(vnew + size_t(n0) * kL + c * T, vn_cur, tid);
    lif_issue_tile_store(spk  + size_t(n0) * kL + c * T, sp_cur, tid);
  }

  lif_wait_async_0();  // drain tail stores (S_ENDPGM would also wait-idle)
}

// ---- host ------------------------------------------------------------------
extern "C" void kernel_launch(void* const* d_in, const int* in_sizes, int n_in,
                              void* d_out, int out_size, void* d_ws, size_t ws_size,
                              hipStream_t stream) {
  (void)in_sizes; (void)n_in; (void)out_size; (void)d_ws; (void)ws_size;

  const float* I = (const float*)d_in[0];
  float* out  = (float*)d_out;
  float* vpre = out;                              // [B,F,L]
  float* vnew = out + size_t(kN) * kL;            // [B,F,L]
  float* spk  = out + 2 * size_t(kN) * kL;        // [B,F,L]

  // 144KB dynamic LDS per block (CDNA5 allows up to 320KB per workgroup).
  (void)hipFuncSetAttribute((const void*)LIFLayer_86260123173247_kernel,
                            hipFuncAttributeMaxDynamicSharedMemorySize,
                            (int)LDS_BYTES);

  LIFLayer_86260123173247_kernel<<<kN / BLOCK, BLOCK, LDS_BYTES, stream>>>(
      I, vpre, vnew, spk);
}